// TransformationWithin_63067299774971
// MI455X (gfx1250) — compile-verified
//
#include <hip/hip_runtime.h>

typedef float v2f __attribute__((ext_vector_type(2)));
typedef float v8f __attribute__((ext_vector_type(8)));

#define RDIM 128
#define LDA  132   // LDS row stride (floats) for the 128x128 A matrix (bank-spread)
#define WTILE 64   // output-column tile
#define LDB  80    // LDS row stride (floats) for the 128x64 B panel (bank-spread)
#define TPB  256

__device__ __forceinline__ v8f wmma_f32_16x16x4(v2f a, v2f b, v8f c) {
  // 8 args: (neg_a, A, neg_b, B, c_mod, C, reuse_a, reuse_b)
  return __builtin_amdgcn_wmma_f32_16x16x4_f32(false, a, false, b, (short)0, c,
                                               false, false);
}

// Async global->LDS panel fetch: 128x64 floats = 2048 x b128 = 8 wave-instrs
// of global_load_async_to_lds_b128 (GV mode), tracked by ASYNCcnt.
__device__ __forceinline__ void async_panel_load(const float* __restrict__ B,
                                                 int ldb, int n0,
                                                 unsigned lds_base, int tid) {
  #pragma unroll
  for (int i = 0; i < 8; ++i) {
    int flat = i * (TPB * 4) + tid * 4;
    int r = flat >> 6, c = flat & 63;
    unsigned loff = lds_base + (unsigned)((r * LDB + c) * 4);
    unsigned long long ga =
        (unsigned long long)(const void*)(B + (size_t)r * (size_t)ldb + n0 + c);
    asm volatile("global_load_async_to_lds_b128 %0, %1, off"
                 :: "v"(loff), "v"(ga) : "memory");
  }
}

__device__ __forceinline__ void wait_async_le8() {
  asm volatile("s_wait_asynccnt 0x8" ::: "memory");
}
__device__ __forceinline__ void wait_async_0() {
  asm volatile("s_wait_asynccnt 0x0" ::: "memory");
}

// ---------------------------------------------------------------------------
// Gauss-Jordan inversion with partial pivoting of three 128x128 matrices.
// One block per matrix. Result: invg[b] = inv(g_b), row-major 128x128.
// ---------------------------------------------------------------------------
#define ITPB 512
__global__ __launch_bounds__(ITPB) void invert128_kernel(
    const float* __restrict__ g0, const float* __restrict__ g1,
    const float* __restrict__ g2, float* __restrict__ invg) {
  __shared__ float As[RDIM * 129];
  __shared__ float Iv[RDIM * 129];
  __shared__ float fac[RDIM];
  __shared__ int   spiv;
  __shared__ float spval;

  const float* G = (blockIdx.x == 0) ? g0 : (blockIdx.x == 1) ? g1 : g2;
  float* out = invg + (size_t)blockIdx.x * RDIM * RDIM;
  const int tid = threadIdx.x;

  for (int idx = tid; idx < RDIM * RDIM; idx += ITPB) {
    int r = idx >> 7, c = idx & 127;
    As[r * 129 + c] = G[idx];
    Iv[r * 129 + c] = (r == c) ? 1.0f : 0.0f;
  }
  __syncthreads();

  for (int k = 0; k < RDIM; ++k) {
    if (tid == 0) {
      int best = k;
      float bv = fabsf(As[k * 129 + k]);
      for (int r = k + 1; r < RDIM; ++r) {
        float av = fabsf(As[r * 129 + k]);
        if (av > bv) { bv = av; best = r; }
      }
      spiv = best;
      spval = As[best * 129 + k];  // pivot value after the swap below
    }
    __syncthreads();
    const int piv = spiv;
    const float rp = 1.0f / spval;
    if (piv != k) {
      for (int c = tid; c < RDIM; c += ITPB) {
        float t1 = As[k * 129 + c]; As[k * 129 + c] = As[piv * 129 + c]; As[piv * 129 + c] = t1;
        float t2 = Iv[k * 129 + c]; Iv[k * 129 + c] = Iv[piv * 129 + c]; Iv[piv * 129 + c] = t2;
      }
    }
    __syncthreads();
    // normalize pivot row; capture elimination factors (disjoint locations)
    for (int c = tid; c < RDIM; c += ITPB) {
      As[k * 129 + c] *= rp;
      Iv[k * 129 + c] *= rp;
    }
    for (int r = tid; r < RDIM; r += ITPB)
      fac[r] = (r == k) ? 0.0f : As[r * 129 + k];
    __syncthreads();
    // eliminate
    for (int idx = tid; idx < RDIM * 256; idx += ITPB) {
      int r = idx >> 8;
      if (r == k) continue;
      int c = idx & 255;
      float f = fac[r];
      if (c < 128) As[r * 129 + c] -= f * As[k * 129 + c];
      else         Iv[r * 129 + (c - 128)] -= f * Iv[k * 129 + (c - 128)];
    }
    __syncthreads();
  }

  for (int idx = tid; idx < RDIM * RDIM; idx += ITPB) {
    int r = idx >> 7, c = idx & 127;
    out[idx] = Iv[r * 129 + c];
  }
}

// ---------------------------------------------------------------------------
// D(128 x ncols) = A(128x128 or its transpose) @ B(128 x ldb).
// Persistent over `ntiles` consecutive 64-column tiles with double-buffered
// LDS panels filled by async global->LDS loads. 8 waves; each wave owns
// 16 rows x 64 cols via 4 WMMA f32 16x16x4 accumulators.
// ---------------------------------------------------------------------------
template <int TRANSA>
__global__ __launch_bounds__(TPB) void gemm_rank_kernel(
    const float* __restrict__ A, const float* __restrict__ Bsrc,
    float* __restrict__ D, int ldb, int ntiles) {
  __shared__ __align__(16) float Asl[RDIM * LDA];
  __shared__ __align__(16) float Stl[2][RDIM * LDB];

  const int tid = threadIdx.x;
  const int tile_base = blockIdx.x * ntiles;

  const unsigned sbase0 = (unsigned)(unsigned long long)(uintptr_t)&Stl[0][0];
  const unsigned sbase1 = (unsigned)(unsigned long long)(uintptr_t)&Stl[1][0];

  // Prefetch first panel asynchronously.
  async_panel_load(Bsrc, ldb, tile_base * WTILE, sbase0, tid);

  // Stage A (128x128) into LDS, float4 per thread x 16 (L2-resident across blocks).
  #pragma unroll
  for (int i = 0; i < 16; ++i) {
    int flat = i * (TPB * 4) + tid * 4;
    int r = flat >> 7, c = flat & 127;
    const float4 av = *reinterpret_cast<const float4*>(A + flat);
    *reinterpret_cast<float4*>(&Asl[r * LDA + c]) = av;
  }

  const int wave = tid >> 5;
  const int lane = tid & 31;
  const int sub  = lane >> 4;   // lane group: 0 or 1
  const int lr   = lane & 15;
  const int m0   = wave * 16;

  for (int t = 0; t < ntiles; ++t) {
    const int cur = t & 1;
    float* St = &Stl[cur][0];
    const int n0 = (tile_base + t) * WTILE;

    // Prefetch next panel into the other buffer while we compute this one.
    const bool have_next = (t + 1) < ntiles;
    if (have_next) {
      async_panel_load(Bsrc, ldb, (tile_base + t + 1) * WTILE,
                       cur ? sbase0 : sbase1, tid);
      wait_async_le8();   // 8 newer instrs outstanding -> this tile's landed
    } else {
      wait_async_0();
    }
    __syncthreads();

    v8f acc[4];
    #pragma unroll
    for (int q = 0; q < 4; ++q)
      #pragma unroll
      for (int e = 0; e < 8; ++e) acc[q][e] = 0.0f;

    #pragma unroll 4
    for (int k = 0; k < RDIM; k += 4) {
      v2f a;
      if (TRANSA) {  // logical A(m,k) = Astored(k,m): cross-row, scalar reads
        a.x = Asl[(k + 2 * sub    ) * LDA + m0 + lr];
        a.y = Asl[(k + 2 * sub + 1) * LDA + m0 + lr];
      } else {
        // (m)*132 + k + 2*sub is even -> 8B aligned: one ds_load_b64
        a = *reinterpret_cast<const v2f*>(&Asl[(m0 + lr) * LDA + k + 2 * sub]);
      }
      #pragma unroll
      for (int q = 0; q < 4; ++q) {
        v2f b;
        b.x = St[(k + 2 * sub    ) * LDB + q * 16 + lr];
        b.y = St[(k + 2 * sub + 1) * LDB + q * 16 + lr];
        acc[q] = wmma_f32_16x16x4(a, b, acc[q]);
      }
    }

    __syncthreads();
    // Stage D back through LDS (reusing the consumed panel) so global stores
    // are 256B-coalesced float4s.
    #pragma unroll
    for (int q = 0; q < 4; ++q)
      #pragma unroll
      for (int r = 0; r < 8; ++r) {
        int row = m0 + r + 8 * sub;   // D VGPR r: M = r (lanes 0-15), r+8 (16-31)
        St[row * LDB + q * 16 + lr] = acc[q][r];
      }
    __syncthreads();

    #pragma unroll
    for (int i = 0; i < 8; ++i) {
      int flat = i * (TPB * 4) + tid * 4;
      int r = flat >> 6, c = flat & 63;
      float4 ov = *reinterpret_cast<float4*>(&St[r * LDB + c]);
      *reinterpret_cast<float4*>(D + (size_t)r * (size_t)ldb + n0 + c) = ov;
    }
    __syncthreads();  // this buffer must be fully read before async reuse
  }
}

// ---------------------------------------------------------------------------
extern "C" void kernel_launch(void* const* d_in, const int* in_sizes, int n_in,
                              void* d_out, int out_size, void* d_ws,
                              size_t ws_size, hipStream_t stream) {
  (void)in_sizes; (void)n_in; (void)out_size; (void)ws_size;

  const float* g0 = (const float*)d_in[0];
  const float* g1 = (const float*)d_in[1];
  const float* g2 = (const float*)d_in[2];
  const float* v0 = (const float*)d_in[3];
  const float* v1 = (const float*)d_in[4];
  const float* v2 = (const float*)d_in[5];
  const float* s0 = (const float*)d_in[6];
  const float* s1 = (const float*)d_in[7];
  const float* s2 = (const float*)d_in[8];

  float* out  = (float*)d_out;
  float* invg = (float*)d_ws;               // 3 * 128 * 128 floats

  const int    VN = 512;
  const size_t NN = (size_t)512 * 512;

  float* outv0 = out;
  float* outs0 = outv0 + (size_t)RDIM * VN;
  float* outv1 = outs0 + (size_t)RDIM * NN;
  float* outs1 = outv1 + (size_t)RDIM * VN;
  float* outv2 = outs1 + (size_t)RDIM * NN;
  float* outs2 = outv2 + (size_t)RDIM * VN;

  // 1) invert the three g matrices (into workspace)
  invert128_kernel<<<3, ITPB, 0, stream>>>(g0, g1, g2, invg);

  // 2) vn = g^T @ v   (transA GEMM, 512 columns -> 8 blocks x 1 tile)
  gemm_rank_kernel<1><<<VN / WTILE, TPB, 0, stream>>>(g0, v0, outv0, VN, 1);
  gemm_rank_kernel<1><<<VN / WTILE, TPB, 0, stream>>>(g1, v1, outv1, VN, 1);
  gemm_rank_kernel<1><<<VN / WTILE, TPB, 0, stream>>>(g2, v2, outv2, VN, 1);

  // 3) sn = inv(g) @ s (4096 tiles -> 1024 persistent blocks x 4 tiles)
  const int SBLK = 1024, STILES = (int)(NN / WTILE) / SBLK;
  gemm_rank_kernel<0><<<SBLK, TPB, 0, stream>>>(
      invg + 0 * RDIM * RDIM, s0, outs0, (int)NN, STILES);
  gemm_rank_kernel<0><<<SBLK, TPB, 0, stream>>>(
      invg + 1 * RDIM * RDIM, s1, outs1, (int)NN, STILES);
  gemm_rank_kernel<0><<<SBLK, TPB, 0, stream>>>(
      invg + 2 * RDIM * RDIM, s2, outs2, (int)NN, STILES);
}